// BlockAttnResQwen_47957604827783
// MI455X (gfx1250) — compile-verified
//
#include <hip/hip_runtime.h>
#include <hip/hip_bf16.h>
#include <cmath>

#ifndef __has_builtin
#define __has_builtin(x) 0
#endif

#define EPS_RMS 1e-6f
#define MAXN 16

// ---- CDNA5 async global->LDS path (guarded; fallback = plain float4 copy) ----
#if defined(__HIP_DEVICE_COMPILE__)
#if __has_builtin(__builtin_amdgcn_global_load_async_to_lds_b128)
#define HAVE_ASYNC_LDS 1
#endif
#endif

#if HAVE_ASYNC_LDS
// Builtin parameter types (from clang diagnostic): int4* in addrspace(1)/(3).
typedef int v4i_t __attribute__((ext_vector_type(4)));
typedef __attribute__((address_space(1))) v4i_t* glb_v4i_t;
typedef __attribute__((address_space(3))) v4i_t* lds_v4i_t;

#if __has_builtin(__builtin_amdgcn_s_wait_asynccnt)
#define WAIT_ASYNC() __builtin_amdgcn_s_wait_asynccnt(0)
#else
#define WAIT_ASYNC() asm volatile("s_wait_asynccnt 0" ::: "memory")
#endif
// 16 bytes per lane: copies from the global address srcGlb to the LDS address dstLds; tracked with ASYNCcnt.
// cpol: 0 = RT (default), 1 = TH_NT (non-temporal; stream past L2 retention).
#define ASYNC_CP16_NT(srcGlb, dstLds)                                           \
  __builtin_amdgcn_global_load_async_to_lds_b128(                               \
      (glb_v4i_t)(srcGlb), (lds_v4i_t)(dstLds), 0, 1)
#define ASYNC_CP16(srcGlb, dstLds)                                              \
  __builtin_amdgcn_global_load_async_to_lds_b128(                               \
      (glb_v4i_t)(srcGlb), (lds_v4i_t)(dstLds), 0, 0)
#else
#define WAIT_ASYNC() ((void)0)
#define ASYNC_CP16_NT(srcGlb, dstLds)                                           \
  do { *(float4*)(dstLds) = *(const float4*)(srcGlb); } while (0)
#define ASYNC_CP16(srcGlb, dstLds)                                              \
  do { *(float4*)(dstLds) = *(const float4*)(srcGlb); } while (0)
#endif

// -----------------------------------------------------------------------------
// Main: one workgroup per flattened (b,t) row. Single pass over `values`:
// async-stage the n rows (+ query) into LDS (104KB of the 320KB/WGP),
// reduce ssq & q.v per row, softmax over n, weighted-sum back out.
// -----------------------------------------------------------------------------
__global__ void __launch_bounds__(256)
route_kernel(const float* __restrict__ values,
             const float* __restrict__ qvec,
             const float* __restrict__ qbias,
             float* __restrict__ routed,
             float* __restrict__ alpha_out,
             int n, int h, float inv_scale, float inv_h)
{
    extern __shared__ float sV[];        // (n+1)*h floats; last row holds query
    __shared__ float sSsq[MAXN];
    __shared__ float sQv[MAXN];

    const int tid = threadIdx.x;
    const int bt  = blockIdx.x;
    const size_t planeStride = (size_t)gridDim.x * (size_t)h;   // BT*h
    const float* __restrict__ rowBase = values + (size_t)bt * (size_t)h;

    // 1) Kick off async copies: n value rows (non-temporal; streamed once)
    //    plus the query vector (regular policy; shared by all workgroups).
    for (int j = 0; j < n; ++j) {
        const float* g = rowBase + (size_t)j * planeStride;
        float* l = sV + (size_t)j * h;
        for (int c = tid * 4; c < h; c += 256 * 4)
            ASYNC_CP16_NT(g + c, l + c);
    }
    {
        float* l = sV + (size_t)n * h;
        for (int c = tid * 4; c < h; c += 256 * 4)
            ASYNC_CP16(qvec + c, l + c);
    }
    if (tid < MAXN) { sSsq[tid] = 0.f; sQv[tid] = 0.f; }
    WAIT_ASYNC();
    __syncthreads();

    // 2) Per-row sum-of-squares and q.v (float4 LDS reads, wave32 reduce).
    const int h4 = h >> 2;
    const float4* __restrict__ q4 = (const float4*)(sV + (size_t)n * h);
    for (int j = 0; j < n; ++j) {
        const float4* __restrict__ v4 = (const float4*)(sV + (size_t)j * h);
        float ssq = 0.f, qv = 0.f;
        for (int c = tid; c < h4; c += 256) {
            float4 v = v4[c];
            float4 q = q4[c];
            ssq = fmaf(v.x, v.x, ssq); ssq = fmaf(v.y, v.y, ssq);
            ssq = fmaf(v.z, v.z, ssq); ssq = fmaf(v.w, v.w, ssq);
            qv  = fmaf(q.x, v.x, qv);  qv  = fmaf(q.y, v.y, qv);
            qv  = fmaf(q.z, v.z, qv);  qv  = fmaf(q.w, v.w, qv);
        }
#pragma unroll
        for (int off = 16; off > 0; off >>= 1) {
            ssq += __shfl_down(ssq, off, 32);
            qv  += __shfl_down(qv,  off, 32);
        }
        if ((tid & 31) == 0) {
            atomicAdd(&sSsq[j], ssq);
            atomicAdd(&sQv[j],  qv);
        }
    }
    __syncthreads();

    // 3) Scores + softmax over the n axis (redundant per thread, n <= 16).
    float a[MAXN];
    float m = -3.402823466e38f;
    for (int j = 0; j < n; ++j) {
        float inv = rsqrtf(sSsq[j] * inv_h + EPS_RMS);   // rsqrt(mean + eps)
        float s = (inv * sQv[j] + qbias[j]) * inv_scale; // (q.keys)/sqrt(h)
        a[j] = s;
        m = fmaxf(m, s);
    }
    float denom = 0.f;
    for (int j = 0; j < n; ++j) { float e = expf(a[j] - m); a[j] = e; denom += e; }
    const float rden = 1.f / denom;
    for (int j = 0; j < n; ++j) a[j] *= rden;

    // 4) routed[bt, :] = sum_n alpha_n * v_n ; alpha_out[bt, :] = alpha.
    float4* __restrict__ out4 = (float4*)(routed + (size_t)bt * (size_t)h);
    for (int c = tid; c < h4; c += 256) {
        float4 acc = make_float4(0.f, 0.f, 0.f, 0.f);
        for (int j = 0; j < n; ++j) {
            float4 v = ((const float4*)(sV + (size_t)j * h))[c];
            acc.x = fmaf(a[j], v.x, acc.x);
            acc.y = fmaf(a[j], v.y, acc.y);
            acc.z = fmaf(a[j], v.z, acc.z);
            acc.w = fmaf(a[j], v.w, acc.w);
        }
        out4[c] = acc;
    }
    if (tid < n) alpha_out[(size_t)bt * (size_t)n + tid] = a[tid];
}

// -----------------------------------------------------------------------------
// Prep: qvec[h] = w_query[position]; qbias[j] = dot(qvec, key_pos_bias[j])
// One workgroup; trivial cost, hoisted out of the 4096 main workgroups.
// -----------------------------------------------------------------------------
__global__ void __launch_bounds__(256)
qprep_kernel(const float* __restrict__ w_query,
             const float* __restrict__ key_pos_bias,
             const int*   __restrict__ position,
             float* __restrict__ qvec,
             float* __restrict__ qbias,
             int n, int h)
{
    __shared__ float red[8];
    const int tid = threadIdx.x;
    const int pos = *position;
    const float* __restrict__ q = w_query + (size_t)pos * (size_t)h;

    for (int i = tid; i < h; i += 256) qvec[i] = q[i];

    for (int j = 0; j < n; ++j) {
        const float* __restrict__ b = key_pos_bias + (size_t)j * (size_t)h;
        float p = 0.f;
        for (int i = tid; i < h; i += 256) p = fmaf(q[i], b[i], p);
#pragma unroll
        for (int off = 16; off > 0; off >>= 1) p += __shfl_down(p, off, 32);
        if ((tid & 31) == 0) red[tid >> 5] = p;
        __syncthreads();
        if (tid == 0) {
            float s = 0.f;
#pragma unroll
            for (int w = 0; w < 8; ++w) s += red[w];
            qbias[j] = s;
        }
        __syncthreads();
    }
}

// -----------------------------------------------------------------------------
extern "C" void kernel_launch(void* const* d_in, const int* in_sizes, int n_in,
                              void* d_out, int out_size, void* d_ws, size_t ws_size,
                              hipStream_t stream)
{
    const float* values = (const float*)d_in[0];
    const float* wq     = (const float*)d_in[1];
    const float* kpb    = (const float*)d_in[2];
    const int*   pos    = (const int*)d_in[3];

    // Shapes: values [n, BT, h]; w_query [16, h]; out = BT*h + BT*n.
    const long long S0 = (long long)in_sizes[0];
    const int h = in_sizes[1] / 16;                       // NUM_BLOCKS == 16
    const long long den = (long long)out_size * (long long)h - S0;
    const int n  = (int)((S0 * (long long)h) / den);      // = 12
    const int BT = (int)(S0 / ((long long)h * (long long)n)); // = 4096

    float* qvec  = (float*)d_ws;          // h floats
    float* qbias = qvec + h;              // n floats (ws needs ~(h+16)*4 B)
    float* routed = (float*)d_out;        // [BT, h]
    float* alpha  = routed + (size_t)BT * (size_t)h;      // [BT, n]

    const float inv_scale = 1.0f / sqrtf((float)h);       // TEMPERATURE == 1
    const float inv_h     = 1.0f / (float)h;

    qprep_kernel<<<1, 256, 0, stream>>>(wq, kpb, pos, qvec, qbias, n, h);

    const size_t shmem = (size_t)(n + 1) * (size_t)h * sizeof(float); // 104 KB
    route_kernel<<<BT, 256, shmem, stream>>>(values, qvec, qbias,
                                             routed, alpha,
                                             n, h, inv_scale, inv_h);
}